// UpdateRule_44727789421163
// MI455X (gfx1250) — compile-verified
//
#include <hip/hip_runtime.h>
#include <math.h>

#define HID 128
#define EDIM 16
#define NEG_SLOPE 0.2f

typedef __attribute__((ext_vector_type(2))) float v2f;
typedef __attribute__((ext_vector_type(8))) float v8f;

// ---------------------------------------------------------------- utilities
__global__ void k_fill(float* __restrict__ p, float v, long n) {
  long i = (long)blockIdx.x * blockDim.x + threadIdx.x;
  if (i < n) p[i] = v;
}

// Sum edge_attr columns into acc[16] (later divided by E).
__global__ void k_ea_mean(const float* __restrict__ ea, float* __restrict__ acc, long E) {
  __shared__ float s[EDIM];
  if (threadIdx.x < EDIM) s[threadIdx.x] = 0.f;
  __syncthreads();
  const int c = threadIdx.x & (EDIM - 1);
  const long rowsPer = blockDim.x / EDIM;
  float local = 0.f;
  for (long e = (long)blockIdx.x * rowsPer + (threadIdx.x >> 4); e < E;
       e += (long)gridDim.x * rowsPer)
    local += ea[e * EDIM + c];
  atomicAdd(&s[c], local);
  __syncthreads();
  if (threadIdx.x < EDIM) atomicAdd(&acc[threadIdx.x], s[threadIdx.x]);
}

// wevec[k] = sum_j We[k][j]*ae[j];  loopae = dot(mean_ea, wevec)
__global__ void k_layer_prep(const float* __restrict__ We, const float* __restrict__ ae,
                             const float* __restrict__ meanacc, long E, int dout,
                             float* __restrict__ wevec, float* __restrict__ loopae) {
  __shared__ float s[EDIM];
  const int k = threadIdx.x;
  if (k < EDIM) {
    float a = 0.f;
    for (int j = 0; j < dout; ++j) a += We[k * dout + j] * ae[j];
    wevec[k] = a;
    s[k] = a;
  }
  __syncthreads();
  if (k == 0) {
    const float inv = 1.f / (float)E;
    float a = 0.f;
    for (int t = 0; t < EDIM; ++t) a += meanacc[t] * inv * s[t];
    *loopae = a;
  }
}

// ---------------------------------------------------------------- WMMA GEMM
// H[N x dout] = X[N x K] * W[K x dout], one wave per 16x16 tile, fp32 WMMA.
// A frag (16x4): lanes 0-15 hold K={k,k+1}, lanes 16-31 hold K={k+2,k+3} for
// row M = lane&15. B frag (4x16) symmetric. D: VGPR r -> M = r + 8*(lane>=16).
__global__ void k_gemm_wmma(const float* __restrict__ X, const float* __restrict__ W,
                            float* __restrict__ H, int Mtiles, int Ntiles, int K, int dout) {
  const int wave = blockIdx.x * (blockDim.x >> 5) + (threadIdx.x >> 5);
  if (wave >= Mtiles * Ntiles) return;  // uniform per wave: EXEC stays all-1s
  const int lane = threadIdx.x & 31;
  const int mt = wave / Ntiles;
  const int nt = wave % Ntiles;
  const int halfsel = lane >> 4;   // 0 or 1
  const int l15 = lane & 15;
  const long arow = (long)mt * 16 + l15;
  const int bcol = nt * 16 + l15;

  v8f c = {};
  for (int k = 0; k < K; k += 4) {
    const int kk = k + (halfsel << 1);
    v2f a, b;
    a[0] = X[arow * K + kk];
    a[1] = X[arow * K + kk + 1];
    b[0] = W[(long)kk * dout + bcol];
    b[1] = W[(long)(kk + 1) * dout + bcol];
    c = __builtin_amdgcn_wmma_f32_16x16x4_f32(false, a, false, b, (short)0, c, false, false);
  }

  const long rbase = (long)mt * 16 + halfsel * 8;
#pragma unroll
  for (int r = 0; r < 8; ++r) H[(rbase + r) * dout + bcol] = c[r];
}

// ---------------------------------------------------------------- node stage
__global__ void k_node_alpha(const float* __restrict__ h, const float* __restrict__ a_s,
                             const float* __restrict__ a_d, float* __restrict__ asrc,
                             float* __restrict__ adst, float* __restrict__ m,
                             float* __restrict__ denom, long N, int C) {
  long i = (long)blockIdx.x * blockDim.x + threadIdx.x;
  if (i >= N) return;
  float s = 0.f, d = 0.f;
  const float* hp = h + i * C;
  for (int c = 0; c < C; ++c) {
    float v = hp[c];
    s += v * a_s[c];
    d += v * a_d[c];
  }
  asrc[i] = s;
  adst[i] = d;
  m[i] = -3.402823466e38f;
  denom[i] = 0.f;
}

// ---------------------------------------------------------------- edge stage
__global__ void k_edge_alpha(const int* __restrict__ src, const int* __restrict__ dst,
                             const float* __restrict__ eattr, const float* __restrict__ wevec,
                             const float* __restrict__ loopae, const float* __restrict__ asrc,
                             const float* __restrict__ adst, float* __restrict__ alpha,
                             float* __restrict__ m, long E, long N) {
  long e = (long)blockIdx.x * blockDim.x + threadIdx.x;
  if (e >= E + N) return;
  int s, d;
  float ae_v;
  if (e < E) {
    s = src[e];
    d = dst[e];
    float acc = 0.f;
    const float* ep = eattr + e * EDIM;
#pragma unroll
    for (int k = 0; k < EDIM; ++k) acc += ep[k] * wevec[k];
    ae_v = acc;
  } else {
    s = d = (int)(e - E);
    ae_v = *loopae;
  }
  float a = asrc[s] + adst[d] + ae_v;
  a = a > 0.f ? a : NEG_SLOPE * a;
  alpha[e] = a;
  __hip_atomic_fetch_max(&m[d], a, __ATOMIC_RELAXED, __HIP_MEMORY_SCOPE_AGENT);
}

__global__ void k_edge_exp(const int* __restrict__ dst, const float* __restrict__ m,
                           float* __restrict__ alpha, float* __restrict__ denom,
                           long E, long N) {
  long e = (long)blockIdx.x * blockDim.x + threadIdx.x;
  if (e >= E + N) return;
  const int d = (e < E) ? dst[e] : (int)(e - E);
  const float ea = __expf(alpha[e] - m[d]);
  alpha[e] = ea;  // in-place: alpha buffer now holds ea
  atomicAdd(&denom[d], ea);
}

// One block per edge; threads cover channels.
__global__ void k_edge_aggr(const int* __restrict__ src, const int* __restrict__ dst,
                            const float* __restrict__ ea, const float* __restrict__ denom,
                            const float* __restrict__ h, float* __restrict__ out,
                            long E, long N, int C) {
  const long e = blockIdx.x;
  if (e >= E + N) return;
  int s, d;
  if (e < E) {
    s = src[e];
    d = dst[e];
  } else {
    s = d = (int)(e - E);
  }
  const float coef = ea[e] / denom[d];
  const int c = threadIdx.x;
  if (c < C) atomicAdd(&out[(long)d * C + c], h[(long)s * C + c] * coef);
}

// ---------------------------------------------------------------- epilogues
__global__ void k_bias_relu(float* __restrict__ out, const float* __restrict__ b, long N, int C) {
  long i = (long)blockIdx.x * blockDim.x + threadIdx.x;
  if (i >= N * C) return;
  float v = out[i] + b[i % C];
  out[i] = fmaxf(v, 0.f);
}

__global__ void k_final(const float* __restrict__ agg, const float* __restrict__ b,
                        const float* __restrict__ x, float* __restrict__ y, long N) {
  long i = (long)blockIdx.x * blockDim.x + threadIdx.x;
  if (i >= N * HID) return;
  y[i] = agg[i] + b[i % HID] + x[i];
}

// ---------------------------------------------------------------- launch
extern "C" void kernel_launch(void* const* d_in, const int* in_sizes, int n_in,
                              void* d_out, int out_size, void* d_ws, size_t ws_size,
                              hipStream_t stream) {
  const float* x = (const float*)d_in[0];
  const int* ei = (const int*)d_in[1];
  const float* eattr = (const float*)d_in[2];

  const long N = in_sizes[0] / HID;
  const long E = in_sizes[2] / EDIM;
  const int* src = ei;
  const int* dst = ei + E;

  // workspace layout (floats)
  float* ws = (float*)d_ws;
  float* hA = ws;                  // N*HID
  float* hB = hA + N * HID;        // N*HID
  float* asrc = hB + N * HID;      // N
  float* adst = asrc + N;          // N
  float* mbuf = adst + N;          // N
  float* denom = mbuf + N;         // N
  float* alpha = denom + N;        // E + N   (reused as ea)
  float* meanacc = alpha + (E + N);  // 16
  float* wevec = meanacc + EDIM;     // 16
  float* loopae = wevec + EDIM;      // 1

  const long EN = E + N;

  // mean(edge_attr) accumulator
  k_fill<<<1, 64, 0, stream>>>(meanacc, 0.f, EDIM);
  k_ea_mean<<<2048, 256, 0, stream>>>(eattr, meanacc, E);

  struct LayerP {
    const float *W, *a_s, *a_d, *We, *ae, *b;
    int din, dout;
  } L[3] = {
      {(const float*)d_in[3], (const float*)d_in[4], (const float*)d_in[5],
       (const float*)d_in[6], (const float*)d_in[7], (const float*)d_in[8], HID, 80},
      {(const float*)d_in[9], (const float*)d_in[10], (const float*)d_in[11],
       (const float*)d_in[12], (const float*)d_in[13], (const float*)d_in[14], 80, 80},
      {(const float*)d_in[15], (const float*)d_in[16], (const float*)d_in[17],
       (const float*)d_in[18], (const float*)d_in[19], (const float*)d_in[20], 80, HID},
  };

  const int edgeBlocks = (int)((EN + 255) / 256);

  for (int li = 0; li < 3; ++li) {
    const float* hin = (li == 0) ? x : hB;
    const int C = L[li].dout;
    const int Mtiles = (int)(N / 16);       // N = 100000 -> 6250 exact
    const int Ntiles = C / 16;              // 80 -> 5, 128 -> 8
    const int tiles = Mtiles * Ntiles;
    const int wavesPerBlock = 8;            // 256 threads

    // h = hin @ W  (fp32 WMMA)
    k_gemm_wmma<<<(tiles + wavesPerBlock - 1) / wavesPerBlock, 256, 0, stream>>>(
        hin, L[li].W, hA, Mtiles, Ntiles, L[li].din, C);

    // node-level scalars + init m/denom
    k_node_alpha<<<(int)((N + 255) / 256), 256, 0, stream>>>(
        hA, L[li].a_s, L[li].a_d, asrc, adst, mbuf, denom, N, C);

    // edge-attr projection for this layer
    k_layer_prep<<<1, 32, 0, stream>>>(L[li].We, L[li].ae, meanacc, E, C, wevec, loopae);

    // output accumulator (hB is dead after GEMM consumed it)
    k_fill<<<(int)((N * C + 255) / 256), 256, 0, stream>>>(hB, 0.f, N * (long)C);

    // segment softmax + aggregation
    k_edge_alpha<<<edgeBlocks, 256, 0, stream>>>(src, dst, eattr, wevec, loopae,
                                                 asrc, adst, alpha, mbuf, E, N);
    k_edge_exp<<<edgeBlocks, 256, 0, stream>>>(dst, mbuf, alpha, denom, E, N);
    k_edge_aggr<<<(int)EN, 128, 0, stream>>>(src, dst, alpha, denom, hA, hB, E, N, C);

    if (li < 2) {
      k_bias_relu<<<(int)((N * C + 255) / 256), 256, 0, stream>>>(hB, L[li].b, N, C);
    } else {
      k_final<<<(int)((N * HID + 255) / 256), 256, 0, stream>>>(hB, L[li].b, x,
                                                                (float*)d_out, N);
    }
  }
}